// DSGraphGLayer_58841051955373
// MI455X (gfx1250) — compile-verified
//
#include <hip/hip_runtime.h>
#include <hip/hip_bf16.h>
#include <stdint.h>

// Problem constants (match reference)
#define Nn   32
#define Mm   2048
#define Dd   128
#define Kk   3
#define Ee   32768
#define MD   (Mm * Dd)          // 262144 columns for the hop GEMM
#define NMD  ((size_t)Nn * Mm * Dd)

typedef float v2f __attribute__((ext_vector_type(2)));
typedef float v8f __attribute__((ext_vector_type(8)));

__device__ __forceinline__ v8f wmma_f32_16x16x4(v2f a, v2f b, v8f c) {
  // D = A(16x4,f32) * B(4x16,f32) + C(16x16,f32); exact fp32 matrix op
  return __builtin_amdgcn_wmma_f32_16x16x4_f32(false, a, false, b, (short)0, c,
                                               false, false);
}

// ---------------------------------------------------------------- setup ----
__global__ void zero_counts_kernel(float* degf, int* fill) {
  int i = blockIdx.x * blockDim.x + threadIdx.x;
  if (i < Mm) { degf[i] = 0.0f; fill[i] = 0; }
}

__global__ void count_kernel(const int* __restrict__ dst, float* degf) {
  int e = blockIdx.x * blockDim.x + threadIdx.x;
  if (e < Ee) atomicAdd(&degf[dst[e]], 1.0f);
}

__global__ void dinv_kernel(const float* __restrict__ degf,
                            float* __restrict__ dinv,
                            float* __restrict__ dinv2) {
  int i = blockIdx.x * blockDim.x + threadIdx.x;
  if (i < Mm) {
    float r = rsqrtf(degf[i] + 1.0f);  // deg = count + self loop
    dinv[i] = r;
    dinv2[i] = r * r;
  }
}

// exclusive scan of 2048 counts -> rowptr[0..2048], one block of 1024 threads
__global__ __launch_bounds__(1024) void scan_kernel(const float* __restrict__ degf,
                                                    int* __restrict__ rowptr) {
  __shared__ int bufA[Mm];
  __shared__ int bufB[Mm];
  int t = threadIdx.x;
  for (int i = t; i < Mm; i += 1024) bufA[i] = (int)degf[i];
  int* src = bufA;
  int* dst = bufB;
  for (int off = 1; off < Mm; off <<= 1) {
    __syncthreads();
    for (int i = t; i < Mm; i += 1024)
      dst[i] = src[i] + ((i >= off) ? src[i - off] : 0);
    int* tmp = src; src = dst; dst = tmp;
  }
  __syncthreads();
  if (t == 0) rowptr[0] = 0;
  for (int i = t; i < Mm; i += 1024) rowptr[i + 1] = src[i];  // inclusive -> +1
}

__global__ void fill_kernel(const int* __restrict__ srcE,
                            const int* __restrict__ dstE,
                            const float* __restrict__ dinv,
                            const int* __restrict__ rowptr,
                            int* __restrict__ fill,
                            int* __restrict__ csr_src,
                            float* __restrict__ csr_coef) {
  int e = blockIdx.x * blockDim.x + threadIdx.x;
  if (e < Ee) {
    int d = dstE[e];
    int s = srcE[e];
    int slot = rowptr[d] + atomicAdd(&fill[d], 1);
    csr_src[slot] = s;
    csr_coef[slot] = dinv[s] * dinv[d];
  }
}

__global__ void bsum_kernel(const float* __restrict__ bs, float* __restrict__ bsum) {
  int e = threadIdx.x;  // 128 threads
  bsum[e] = bs[e] + bs[Dd + e] + bs[2 * Dd + e] + bs[3 * Dd + e];
}

// ------------------------------------------------------------- hop GEMM ----
// Out[n, col] = sum_j C[n,j] * In[j, col], col in [0, M*D). One wave owns a
// 16-wide column strip and ALL 32 rows -> in-place safe (In may == Out).
__global__ __launch_bounds__(256) void hop_kernel(const float* __restrict__ In,
                                                  float* __restrict__ Out,
                                                  const float* __restrict__ C) {
  __shared__ float Cs[32 * 32];
  int t = threadIdx.x;
  for (int i = t; i < 1024; i += 256) Cs[i] = C[i];
  __syncthreads();

  int wave = t >> 5;
  int lane = t & 31;
  size_t colbase = ((size_t)blockIdx.x * 8 + wave) * 16;

  int hlf  = lane >> 4;          // 0: lanes 0-15, 1: lanes 16-31
  int l15  = lane & 15;

  // B fragments: 8 frags of 4 rows each cover K(j)=0..31 for 16 columns
  v2f bf[8];
#pragma unroll
  for (int kk = 0; kk < 8; kk++) {
    int j0 = kk * 4 + hlf * 2;
    bf[kk].x = In[(size_t)j0 * MD + colbase + l15];
    bf[kk].y = In[(size_t)(j0 + 1) * MD + colbase + l15];
  }

  v8f acc0 = {0.f, 0.f, 0.f, 0.f, 0.f, 0.f, 0.f, 0.f};
  v8f acc1 = acc0;
#pragma unroll
  for (int kk = 0; kk < 8; kk++) {
    int kb = kk * 4 + hlf * 2;
    v2f a0, a1;
    a0.x = Cs[l15 * 32 + kb];
    a0.y = Cs[l15 * 32 + kb + 1];
    a1.x = Cs[(l15 + 16) * 32 + kb];
    a1.y = Cs[(l15 + 16) * 32 + kb + 1];
    acc0 = wmma_f32_16x16x4(a0, bf[kk], acc0);   // out rows 0..15
    acc1 = wmma_f32_16x16x4(a1, bf[kk], acc1);   // out rows 16..31
  }

  int crow = hlf * 8;
#pragma unroll
  for (int r = 0; r < 8; r++) {
    Out[(size_t)(crow + r) * MD + colbase + l15]      = acc0[r];
    Out[(size_t)(crow + r + 16) * MD + colbase + l15] = acc1[r];
  }
}

// ------------------------------------------------------------ main GEMM ----
// Z[r, e] (+)= sum_d Y[r, d] * W[d, e];  r in [0, 65536), d,e in [0,128).
// Block = 8 waves; each wave does a 16x128 strip; W cached transposed in LDS.
__global__ __launch_bounds__(256) void gemm_kernel(const float* __restrict__ Y,
                                                   const float* __restrict__ W,
                                                   float* __restrict__ Z,
                                                   int accumulate) {
  __shared__ float WT[Dd * Dd];  // WT[e*128 + d], 64 KB
  int t = threadIdx.x;
  for (int i = t; i < (Dd * Dd) / 4; i += 256) {
    int base = i * 4;
    int d = base >> 7;
    int e = base & 127;
    float4 w = ((const float4*)W)[i];
    WT[(e + 0) * Dd + d] = w.x;
    WT[(e + 1) * Dd + d] = w.y;
    WT[(e + 2) * Dd + d] = w.z;
    WT[(e + 3) * Dd + d] = w.w;
  }
  __syncthreads();

  int wave = t >> 5;
  int lane = t & 31;
  int hlf  = lane >> 4;
  int l15  = lane & 15;
  size_t rowbase = (size_t)blockIdx.x * 128 + wave * 16;

  v8f acc[8];
  int crow = hlf * 8;
#pragma unroll
  for (int c = 0; c < 8; c++) {
    if (accumulate) {
#pragma unroll
      for (int r = 0; r < 8; r++)
        acc[c][r] = Z[(rowbase + crow + r) * Dd + c * 16 + l15];
    } else {
      acc[c] = (v8f){0.f, 0.f, 0.f, 0.f, 0.f, 0.f, 0.f, 0.f};
    }
  }

  for (int k = 0; k < Dd; k += 4) {
    int kb = k + hlf * 2;
    const float* ap = Y + (rowbase + l15) * Dd + kb;
    v2f a;
    a.x = ap[0];
    a.y = ap[1];
#pragma unroll
    for (int c = 0; c < 8; c++) {
      int col = c * 16 + l15;
      v2f b;
      b.x = WT[col * Dd + kb];
      b.y = WT[col * Dd + kb + 1];
      acc[c] = wmma_f32_16x16x4(a, b, acc[c]);
    }
  }

#pragma unroll
  for (int c = 0; c < 8; c++)
#pragma unroll
    for (int r = 0; r < 8; r++)
      Z[(rowbase + crow + r) * Dd + c * 16 + l15] = acc[c][r];
}

// ------------------------------------------------------------- gather ------
// Out[:, m, :] = sum_{j in bin(m)} coef[j] * Z[:, src[j], :]
//              + dinv2[m] * Z[:, m, :] + bsum[:]
// One block per m; 256 threads hold the 32x128 tile (16 floats each).
__global__ __launch_bounds__(256) void gather_kernel(const float* __restrict__ Z,
                                                     const int* __restrict__ rowptr,
                                                     const int* __restrict__ csr_src,
                                                     const float* __restrict__ csr_coef,
                                                     const float* __restrict__ dinv2,
                                                     const float* __restrict__ bsum,
                                                     float* __restrict__ Out) {
  int m = blockIdx.x;
  int t = threadIdx.x;
  float acc[16];
#pragma unroll
  for (int q = 0; q < 16; q++) acc[q] = 0.0f;

  int beg = rowptr[m];
  int end = rowptr[m + 1];
  for (int j = beg; j < end; j++) {
    int s = csr_src[j];
    float w = csr_coef[j];
    const float* zp = Z + (size_t)s * Dd;
#pragma unroll
    for (int q = 0; q < 16; q++) {
      int f = q * 256 + t;
      int n = f >> 7;
      int e = f & 127;
      acc[q] += w * zp[(size_t)n * MD + e];   // z resident in L2 (32 MB)
    }
  }

  float w2 = dinv2[m];
  const float* zm = Z + (size_t)m * Dd;
#pragma unroll
  for (int q = 0; q < 16; q++) {
    int f = q * 256 + t;
    int n = f >> 7;
    int e = f & 127;
    Out[(size_t)n * MD + (size_t)m * Dd + e] =
        acc[q] + w2 * zm[(size_t)n * MD + e] + bsum[e];
  }
}

// ---------------------------------------------------------------------------
extern "C" void kernel_launch(void* const* d_in, const int* in_sizes, int n_in,
                              void* d_out, int out_size, void* d_ws, size_t ws_size,
                              hipStream_t stream) {
  const float* x    = (const float*)d_in[0];   // [32, 2048, 128]
  const int*   adj  = (const int*)d_in[1];     // [2, E]; row0=src, row1=dst
  const float* cadj = (const float*)d_in[2];   // [3, 32, 32]
  const float* Ws   = (const float*)d_in[3];   // [4, 128, 128]
  const float* bs   = (const float*)d_in[4];   // [4, 128]
  float*       out  = (float*)d_out;           // [32, 2048, 128]

  // workspace layout
  char* p = (char*)d_ws;
  float* y        = (float*)p; p += NMD * sizeof(float);       // 32 MB
  float* z        = (float*)p; p += NMD * sizeof(float);       // 32 MB
  float* degf     = (float*)p; p += Mm * sizeof(float);
  float* dinv     = (float*)p; p += Mm * sizeof(float);
  float* dinv2v   = (float*)p; p += Mm * sizeof(float);
  int*   rowptr   = (int*)p;   p += (Mm + 64) * sizeof(int);
  int*   fill     = (int*)p;   p += Mm * sizeof(int);
  int*   csr_src  = (int*)p;   p += Ee * sizeof(int);
  float* csr_coef = (float*)p; p += Ee * sizeof(float);
  float* bsum     = (float*)p; p += Dd * sizeof(float);

  const int* srcE = adj;
  const int* dstE = adj + Ee;

  // --- normalization + CSR build (tiny) ---
  zero_counts_kernel<<<(Mm + 255) / 256, 256, 0, stream>>>(degf, fill);
  count_kernel<<<Ee / 256, 256, 0, stream>>>(dstE, degf);
  dinv_kernel<<<Mm / 256, 256, 0, stream>>>(degf, dinv, dinv2v);
  scan_kernel<<<1, 1024, 0, stream>>>(degf, rowptr);
  fill_kernel<<<Ee / 256, 256, 0, stream>>>(srcE, dstE, dinv, rowptr, fill,
                                            csr_src, csr_coef);
  bsum_kernel<<<1, Dd, 0, stream>>>(bs, bsum);

  // --- z = x@W0 + (C0 x)@W1 + (C1 C0 x)@W2 + (C2 C1 C0 x)@W3 ---
  const int gemm_blocks = (Nn * Mm) / 128;   // 512
  const int hop_blocks  = MD / (8 * 16);     // 2048
  gemm_kernel<<<gemm_blocks, 256, 0, stream>>>(x, Ws + 0 * Dd * Dd, z, 0);
  hop_kernel<<<hop_blocks, 256, 0, stream>>>(x, y, cadj + 0 * Nn * Nn);
  gemm_kernel<<<gemm_blocks, 256, 0, stream>>>(y, Ws + 1 * Dd * Dd, z, 1);
  hop_kernel<<<hop_blocks, 256, 0, stream>>>(y, y, cadj + 1 * Nn * Nn);
  gemm_kernel<<<gemm_blocks, 256, 0, stream>>>(y, Ws + 2 * Dd * Dd, z, 1);
  hop_kernel<<<hop_blocks, 256, 0, stream>>>(y, y, cadj + 2 * Nn * Nn);
  gemm_kernel<<<gemm_blocks, 256, 0, stream>>>(y, Ws + 3 * Dd * Dd, z, 1);

  // --- Lx = P (x)_m z + bsum ---
  gather_kernel<<<Mm, 256, 0, stream>>>(z, rowptr, csr_src, csr_coef, dinv2v,
                                        bsum, out);
}